// kernel_generated_2_21887153341322
// MI455X (gfx1250) — compile-verified
//
#include <hip/hip_runtime.h>

// CDNA5 (gfx1250) fused grouped-conv kernel using fp32 WMMA (V_WMMA_F32_16X16X4_F32).
//
// Roofline: ~14.5 MFLOP, ~1.2 MB traffic -> launch-latency bound on MI455X
// (23.3 TB/s HBM). So: one fused kernel, intermediate (16x8x14x14 fp32) in LDS,
// both contractions on the WMMA pipe at full fp32 precision.
//
// Decomposition: 14 blocks, one per H-row m (conv1 halo along H -> global x;
// conv2 halo along W -> inside one m-row's LDS slice).
//   Phase 1 (conv1): 8 waves, one per group o. (16k x 192) @ (192 x 16n),
//                    48 chained K=4 WMMAs. D -> LDS out1[k][o][n] (8 KB).
//   Phase 2 (conv2): 32 tasks (16 o2 x 2 M-tiles of i=32). (16 x 24) @ (24 x 16n),
//                    6 chained WMMAs, B from LDS, fp32 stores to global.
//
// Contraction ordering is chosen j-major (c' = j*64 + i for conv1, c' = t*8 + j
// for conv2) so every K=4 WMMA slice has a wave-uniform tap: no per-lane div/mod
// in the inner loops, just constant-stride address increments around each WMMA.

typedef __attribute__((ext_vector_type(2))) float v2f;
typedef __attribute__((ext_vector_type(8))) float v8f;

__global__ __launch_bounds__(256, 1) void fused_grouped_conv_wmma(
    const float* __restrict__ x,   // (64*8, 14, 14): channel = i*8 + o
    const float* __restrict__ w1,  // (16, 64, 3)
    const float* __restrict__ w2,  // (32, 8, 3)
    float* __restrict__ out)       // (512, 196):   channel = i*16 + o2
{
  __shared__ float smem[16 * 8 * 16];  // out1[k][o][n] for this m-row (8 KB)

  const int m      = blockIdx.x;       // H row, 0..13
  const int tid    = threadIdx.x;
  const int wave   = tid >> 5;         // 0..7  (wave32)
  const int lane   = tid & 31;
  const int laneLo = lane & 15;        // M (A/D rows) or N (B/D cols)
  const int hi     = lane >> 4;        // 0 -> K 0/1 half, 1 -> K 2/3 half
  const int n      = laneLo;           // W column (valid if < 14)

  const bool colOk = n < 14;
  const int  naddr = colOk ? n : 0;    // clamped address, mask the value instead

  // ---------------- Phase 1: conv1, one 16x16 tile per wave (group o = wave) ----
  {
    const int o = wave;
    v8f acc = {};
#pragma unroll
    for (int j = 0; j < 3; ++j) {                 // H tap (uniform per slice)
      const int  mj    = m + j - 1;               // position in rolled tensor
      const bool rowOk = (unsigned)mj < 14u;      // wave-uniform halo validity
      int msrc = mj - 1;                          // undo roll(+1): (m+j-2) mod 14
      if (msrc < 0) msrc += 14;
      const float* xb = x + o * 196 + msrc * 14 + naddr;  // + i*1568 per channel
#pragma unroll 4
      for (int ii = 0; ii < 16; ++ii) {           // 16 K=4 slices over i=0..63
        const int i0 = ii * 4 + hi * 2;           // channel for b.x (b.y: i0+1)
        v2f a, b;
        const int aoff = laneLo * 192 + i0 * 3 + j;
        a.x = w1[aoff];                           // w1[k][i0][j]
        a.y = w1[aoff + 3];                       // w1[k][i0+1][j]
        const float b0 = xb[i0 * 1568];           // channel stride 8*196
        const float b1 = xb[(i0 + 1) * 1568];
        const bool v = rowOk && colOk;
        b.x = v ? b0 : 0.0f;
        b.y = v ? b1 : 0.0f;
        acc = __builtin_amdgcn_wmma_f32_16x16x4_f32(
            false, a, false, b, (short)0, acc, false, false);
      }
    }
    // D tile -> LDS: VGPR r holds row M = r + hi*8, column N = laneLo
#pragma unroll
    for (int r = 0; r < 8; ++r) {
      const int k = r + hi * 8;                   // conv1 output channel 0..15
      smem[(k * 8 + o) * 16 + n] = acc[r];        // cols 14,15 are exact zeros
    }
  }

  __syncthreads();

  // ---------------- Phase 2: conv2, 32 tile-tasks over 8 waves ------------------
  for (int task = wave; task < 32; task += 8) {
    const int o2  = task >> 1;                    // conv1 channel 0..15
    const int mt  = task & 1;                     // which 16-row half of i=0..31
    const int row = mt * 16 + laneLo;             // output channel i
    v8f acc = {};
#pragma unroll
    for (int t = 0; t < 3; ++t) {                 // W tap (uniform per slice)
      const int  nn  = n + t - 1;
      const bool v   = (unsigned)nn < 14u;        // per-lane column validity
      const int  nns = v ? nn : 0;
      const float* sb = smem + o2 * 128 + nns;    // + j*16 per group channel
#pragma unroll
      for (int jj = 0; jj < 2; ++jj) {            // 2 K=4 slices over j=0..7
        const int j0 = jj * 4 + hi * 2;
        v2f a, b;
        const int aoff = row * 24 + j0 * 3 + t;
        a.x = w2[aoff];                           // w2[i][j0][t]
        a.y = w2[aoff + 3];                       // w2[i][j0+1][t]
        const float b0 = sb[j0 * 16];
        const float b1 = sb[(j0 + 1) * 16];
        b.x = v ? b0 : 0.0f;
        b.y = v ? b1 : 0.0f;
        acc = __builtin_amdgcn_wmma_f32_16x16x4_f32(
            false, a, false, b, (short)0, acc, false, false);
      }
    }
    if (n < 14) {                                 // predicated stores (post-WMMA)
#pragma unroll
      for (int r = 0; r < 8; ++r) {
        const int i_out = mt * 16 + r + hi * 8;   // 0..31
        out[(i_out * 16 + o2) * 196 + m * 14 + n] = acc[r];
      }
    }
  }
}

extern "C" void kernel_launch(void* const* d_in, const int* in_sizes, int n_in,
                              void* d_out, int out_size, void* d_ws, size_t ws_size,
                              hipStream_t stream) {
  const float* x  = (const float*)d_in[0];   // (1,512,14,14) fp32
  const float* w1 = (const float*)d_in[1];   // (16,64,3)     fp32
  const float* w2 = (const float*)d_in[2];   // (32,8,3)      fp32
  float* out = (float*)d_out;                // (1,512,14,14) fp32

  fused_grouped_conv_wmma<<<dim3(14), dim3(256), 0, stream>>>(x, w1, w2, out);
}